// HierarchicalLoss_738734374979
// MI455X (gfx1250) — compile-verified
//
#include <hip/hip_runtime.h>
#include <math.h>

// Problem constants (from reference): B=256 batch, T=16, S=32, D=768, fp32.
#define BB 256
#define TT 16
#define SS 32
#define DD 768

typedef float v2f __attribute__((ext_vector_type(2)));
typedef float v8f __attribute__((ext_vector_type(8)));

// ---------------------------------------------------------------------------
// One wave computes a 16x16 fp32 tile of A(16xK, row-major) x B(16xK, row-major)^T
// using V_WMMA_F32_16X16X4_F32. A/B operand layout (ISA 7.12.2, 32-bit A 16x4):
//   lane l: M (or N) = l%16 ; VGPR r holds K = r + 2*(l/16)
// -> per K-step of 4, each lane loads one contiguous float2 from row-major data.
// ---------------------------------------------------------------------------
__device__ inline v8f wmma_tile_f32(const float* __restrict__ A,
                                    const float* __restrict__ Bm,
                                    int lda, int ldb, int kLen) {
  const int lane = threadIdx.x & 31;
  const int half = lane >> 4;   // 0: lanes 0-15, 1: lanes 16-31
  const int lo   = lane & 15;
  const float* ap = A  + lo * lda + 2 * half;
  const float* bp = Bm + lo * ldb + 2 * half;
  v8f acc = {};
#pragma unroll 8
  for (int k = 0; k < kLen; k += 4) {
    v2f a = *(const v2f*)(ap + k);
    v2f b = *(const v2f*)(bp + k);
    acc = __builtin_amdgcn_wmma_f32_16x16x4_f32(
        /*neg_a=*/false, a, /*neg_b=*/false, b,
        /*c_mod=*/(short)0, acc, /*reuse_a=*/false, /*reuse_b=*/false);
  }
  return acc;
}

// C/D layout (ISA 7.12.2): VGPR r, lanes 0-15 -> M=r, lanes 16-31 -> M=r+8; N=lane%16
__device__ inline void wmma_store_tile(float* __restrict__ out, int ldo,
                                       v8f acc, float scale) {
  const int lane = threadIdx.x & 31;
  const int half = lane >> 4;
  const int lo   = lane & 15;
#pragma unroll
  for (int r = 0; r < 8; ++r)
    out[(r + 8 * half) * ldo + lo] = scale * acc[r];
}

// ---------------------------------------------------------------------------
// K1: tbar[c,d] = mean_s text_local[c,s,d]     grid=256, block=256
// ---------------------------------------------------------------------------
__global__ void k_mean_text(const float* __restrict__ tl, float* __restrict__ tbar) {
  const int c = blockIdx.x;
  for (int d = threadIdx.x; d < DD; d += blockDim.x) {
    float s = 0.0f;
#pragma unroll 4
    for (int ss = 0; ss < SS; ++ss)
      s += tl[(c * SS + ss) * DD + d];
    tbar[c * DD + d] = s * (1.0f / (float)SS);
  }
}

// ---------------------------------------------------------------------------
// K2: per-batch paired sim (16x32, K=768) via WMMA -> time weights -> vbar.
// grid=256, block=64 (2 waves: wave w covers text columns [16w,16w+16))
// ---------------------------------------------------------------------------
__global__ void k_local_weights_vbar(const float* __restrict__ vl,
                                     const float* __restrict__ tl,
                                     const float* __restrict__ temp,
                                     float* __restrict__ vbar) {
  __shared__ float sim[TT * SS];   // scaled paired similarities
  __shared__ float tw[TT];         // row max, then softmax weights
  const int b = blockIdx.x;
  const float scale = __expf(temp[0]);

  const int wave = threadIdx.x >> 5;
  const float* Ab = vl + (size_t)b * TT * DD;                   // 16 x 768
  const float* Bb = tl + (size_t)b * SS * DD + (wave * 16) * DD; // 16 x 768 slice
  v8f acc = wmma_tile_f32(Ab, Bb, DD, DD, DD);
  {
    const int lane = threadIdx.x & 31;
    const int half = lane >> 4;
    const int lo   = lane & 15;
#pragma unroll
    for (int r = 0; r < 8; ++r)
      sim[(r + 8 * half) * SS + wave * 16 + lo] = scale * acc[r];
  }
  __syncthreads();

  if (threadIdx.x < TT) {
    float m = -INFINITY;
    for (int s = 0; s < SS; ++s) m = fmaxf(m, sim[threadIdx.x * SS + s]);
    tw[threadIdx.x] = m;                       // max_s (scaled sim)
  }
  __syncthreads();

  if (threadIdx.x == 0) {                      // softmax over T (tiny, serial)
    float mm = -INFINITY;
    for (int t = 0; t < TT; ++t) mm = fmaxf(mm, tw[t]);
    float e[TT], ssum = 0.0f;
    for (int t = 0; t < TT; ++t) { e[t] = __expf(tw[t] - mm); ssum += e[t]; }
    const float inv = 1.0f / ssum;
    for (int t = 0; t < TT; ++t) tw[t] = e[t] * inv;
  }
  __syncthreads();

  for (int d = threadIdx.x; d < DD; d += blockDim.x) {
    float s = 0.0f;
#pragma unroll
    for (int t = 0; t < TT; ++t)
      s += tw[t] * vl[((size_t)b * TT + t) * DD + d];
    vbar[b * DD + d] = s;
  }
}

// ---------------------------------------------------------------------------
// K3/K4: C[256x256] = scale * A[256x768] x B[256x768]^T  (fp32 WMMA)
// grid=64, block=128 (4 waves, one 16x16 tile each; 256 tiles total)
// ---------------------------------------------------------------------------
__global__ void k_gemm256_nt(const float* __restrict__ A,
                             const float* __restrict__ Bm,
                             const float* __restrict__ temp,
                             float* __restrict__ C) {
  const float scale = __expf(temp[0]);
  const int wave = threadIdx.x >> 5;
  const int tile = blockIdx.x * 4 + wave;
  const int mt = (tile >> 4) << 4;
  const int nt = (tile & 15) << 4;
  v8f acc = wmma_tile_f32(A + mt * DD, Bm + nt * DD, DD, DD, DD);
  wmma_store_tile(C + mt * BB + nt, BB, acc, scale);
}

// ---------------------------------------------------------------------------
// K5: per-index i: row-lse of gsim, col-lse of gsim, row-lse of nfull.
// grid=256, block=256
// ---------------------------------------------------------------------------
__device__ float lse256(float v, float* sm) {
  const int tid = threadIdx.x;
  sm[tid] = v; __syncthreads();
  for (int s = 128; s > 0; s >>= 1) {
    if (tid < s) sm[tid] = fmaxf(sm[tid], sm[tid + s]);
    __syncthreads();
  }
  const float mx = sm[0]; __syncthreads();
  sm[tid] = __expf(v - mx); __syncthreads();
  for (int s = 128; s > 0; s >>= 1) {
    if (tid < s) sm[tid] += sm[tid + s];
    __syncthreads();
  }
  const float r = mx + __logf(sm[0]); __syncthreads();
  return r;
}

__global__ void k_row_losses(const float* __restrict__ gsim,
                             const float* __restrict__ nfull,
                             float* __restrict__ rl_v2t,
                             float* __restrict__ rl_t2v,
                             float* __restrict__ rl_loc) {
  __shared__ float sm[256];
  const int i = blockIdx.x;
  const int tid = threadIdx.x;

  const float diag_g = gsim[i * BB + i];

  float lse_row = lse256(gsim[i * BB + tid], sm);     // v2t: softmax along axis=1
  if (tid == 0) rl_v2t[i] = lse_row - diag_g;

  float lse_col = lse256(gsim[tid * BB + i], sm);     // t2v: softmax along axis=0
  if (tid == 0) rl_t2v[i] = lse_col - diag_g;

  float lse_loc = lse256(nfull[i * BB + tid], sm);    // local: pos = diagonal
  if (tid == 0) rl_loc[i] = lse_loc - nfull[i * BB + i];
}

// ---------------------------------------------------------------------------
// K6: out = 0.6*0.5*(mean v2t + mean t2v) + 0.4*mean local
// ---------------------------------------------------------------------------
__global__ void k_finalize(const float* __restrict__ rl_v2t,
                           const float* __restrict__ rl_t2v,
                           const float* __restrict__ rl_loc,
                           float* __restrict__ out) {
  __shared__ float sm[256];
  const int tid = threadIdx.x;
  float x = 0.3f * (rl_v2t[tid] + rl_t2v[tid]) + 0.4f * rl_loc[tid];
  sm[tid] = x; __syncthreads();
  for (int s = 128; s > 0; s >>= 1) {
    if (tid < s) sm[tid] += sm[tid + s];
    __syncthreads();
  }
  if (tid == 0) out[0] = sm[0] * (1.0f / (float)BB);
}

// ---------------------------------------------------------------------------
extern "C" void kernel_launch(void* const* d_in, const int* in_sizes, int n_in,
                              void* d_out, int out_size, void* d_ws, size_t ws_size,
                              hipStream_t stream) {
  const float* vg   = (const float*)d_in[0];  // [256,768]
  const float* tg   = (const float*)d_in[1];  // [256,768]
  const float* vl   = (const float*)d_in[2];  // [256,16,768]
  const float* tl   = (const float*)d_in[3];  // [256,32,768]
  const float* temp = (const float*)d_in[4];  // [1]
  float* out = (float*)d_out;

  float* ws     = (float*)d_ws;
  float* tbar   = ws;               // 256*768
  float* vbar   = ws + 196608;      // 256*768
  float* gsim   = ws + 393216;      // 256*256
  float* nfull  = ws + 458752;      // 256*256
  float* rl_v2t = ws + 524288;      // 256
  float* rl_t2v = ws + 524544;      // 256
  float* rl_loc = ws + 524800;      // 256  (total ~2.1 MB)

  k_mean_text<<<256, 256, 0, stream>>>(tl, tbar);
  k_local_weights_vbar<<<256, 64, 0, stream>>>(vl, tl, temp, vbar);
  k_gemm256_nt<<<64, 128, 0, stream>>>(vg, tg, temp, gsim);       // global sim
  k_gemm256_nt<<<64, 128, 0, stream>>>(vbar, tbar, temp, nfull);  // collapsed neg_full
  k_row_losses<<<256, 256, 0, stream>>>(gsim, nfull, rl_v2t, rl_t2v, rl_loc);
  k_finalize<<<1, 256, 0, stream>>>(rl_v2t, rl_t2v, rl_loc, out);
}